// BaselineGraphSAGE_49452253446301
// MI455X (gfx1250) — compile-verified
//
#include <hip/hip_runtime.h>
#include <hip/hip_bf16.h>

#define DIM 128

typedef __attribute__((ext_vector_type(2))) float v2f;
typedef __attribute__((ext_vector_type(8))) float v8f;

// ---------------------------------------------------------------------------
// Zero a float buffer (grid-stride).
// ---------------------------------------------------------------------------
__global__ void sage_zero_f32(float* __restrict__ p, long long n) {
  long long i = (long long)blockIdx.x * blockDim.x + threadIdx.x;
  long long stride = (long long)gridDim.x * blockDim.x;
  for (; i < n; i += stride) p[i] = 0.0f;
}

// ---------------------------------------------------------------------------
// Degree count: cnt[dst[e]] += 1  (f32 so the GEMM kernel reads it directly).
// ---------------------------------------------------------------------------
__global__ void sage_degree(const int* __restrict__ dst, float* __restrict__ cnt, int E) {
  int i = blockIdx.x * blockDim.x + threadIdx.x;
  int stride = gridDim.x * blockDim.x;
  for (; i < E; i += stride) atomicAdd(&cnt[dst[i]], 1.0f);
}

// ---------------------------------------------------------------------------
// Scatter-add: one edge per wave32; each lane moves 4 contiguous floats
// (float4 gather from feat[src], 4 global f32 atomics into agg[dst]).
// feat rows (51.2 MB) and agg rows (51.2 MB) are L2-resident on MI455X
// (192 MB L2) so this is bound by L2 atomic throughput, not HBM.
// ---------------------------------------------------------------------------
__global__ void sage_scatter(const float* __restrict__ feat,
                             const int* __restrict__ src,
                             const int* __restrict__ dst,
                             float* __restrict__ agg, int E) {
  const int lane = threadIdx.x & 31;
  const int e = blockIdx.x * (blockDim.x >> 5) + (threadIdx.x >> 5);
  if (e >= E) return;
  const int s = src[e];
  const int d = dst[e];
  const float4 v = reinterpret_cast<const float4*>(feat + (size_t)s * DIM)[lane];
  float* out = agg + (size_t)d * DIM + lane * 4;
  atomicAdd(out + 0, v.x);
  atomicAdd(out + 1, v.y);
  atomicAdd(out + 2, v.z);
  atomicAdd(out + 3, v.w);
}

// ---------------------------------------------------------------------------
// Fused SAGEConv GEMM (full f32 precision via V_WMMA_F32_16X16X4_F32):
//   out[r,:] = (agg[r,:] / max(cnt[r],1)) @ Wl  +  xin[r,:] @ Wr  +  bias
// Block = 256 threads = 8 waves. Block handles a 16-row stripe; wave w
// produces the 16x16 output tile at columns [16w, 16w+16). K = 128 is
// consumed in 32 steps of K=4; both GEMMs accumulate into one C tile.
//
// Per-lane WMMA layout (ISA 7.12.2, 32-bit operands):
//   A: row M = lane%16, K-pair = {2*(lane/16), 2*(lane/16)+1} (+4*step)
//   B: col N = lane%16, same K-pair
//   C: VGPR r -> row r + 8*(lane/16), col = lane%16
// ---------------------------------------------------------------------------
__global__ __launch_bounds__(256)
void sage_fused_gemm(const float* __restrict__ agg,
                     const float* __restrict__ cnt,
                     const float* __restrict__ xin,
                     const float* __restrict__ Wl,
                     const float* __restrict__ Wr,
                     const float* __restrict__ bias,
                     float* __restrict__ out,
                     int do_relu) {
  const int lane = threadIdx.x & 31;
  const int wave = threadIdx.x >> 5;
  const int m    = lane & 15;   // A row / B col / C col within tile
  const int half = lane >> 4;
  const int row0 = blockIdx.x * 16;
  const int col  = wave * 16 + m;

  const float* aggRow = agg + (size_t)(row0 + m) * DIM;
  const float* xRow   = xin + (size_t)(row0 + m) * DIM;
  const float  inv    = 1.0f / fmaxf(cnt[row0 + m], 1.0f);

  v8f c = {};
#pragma unroll
  for (int s = 0; s < 32; ++s) {
    const int kk = 4 * s + 2 * half;
    v2f a, b;
    // left GEMM: normalized aggregate @ Wl
    a.x = aggRow[kk] * inv;
    a.y = aggRow[kk + 1] * inv;
    b.x = Wl[(size_t)kk * DIM + col];
    b.y = Wl[(size_t)(kk + 1) * DIM + col];
    c = __builtin_amdgcn_wmma_f32_16x16x4_f32(false, a, false, b, (short)0, c,
                                              false, false);
    // right GEMM: self features @ Wr
    a.x = xRow[kk];
    a.y = xRow[kk + 1];
    b.x = Wr[(size_t)kk * DIM + col];
    b.y = Wr[(size_t)(kk + 1) * DIM + col];
    c = __builtin_amdgcn_wmma_f32_16x16x4_f32(false, a, false, b, (short)0, c,
                                              false, false);
  }

  const float bv = bias[col];
#pragma unroll
  for (int r = 0; r < 8; ++r) {
    float v = c[r] + bv;
    if (do_relu) v = fmaxf(v, 0.0f);
    out[(size_t)(row0 + r + 8 * half) * DIM + col] = v;
  }
}

// ---------------------------------------------------------------------------
// Host-side launch: layer1 (zero, degree, scatter(x), gemm+relu -> h),
// layer2 (zero agg, scatter(h), gemm -> out). All on `stream`, scratch in ws.
// ---------------------------------------------------------------------------
extern "C" void kernel_launch(void* const* d_in, const int* in_sizes, int n_in,
                              void* d_out, int out_size, void* d_ws, size_t ws_size,
                              hipStream_t stream) {
  const float* x   = (const float*)d_in[0];
  const int*  edge = (const int*)d_in[1];
  const float* W1l = (const float*)d_in[2];
  const float* b1  = (const float*)d_in[3];
  const float* W1r = (const float*)d_in[4];
  const float* W2l = (const float*)d_in[5];
  const float* b2  = (const float*)d_in[6];
  const float* W2r = (const float*)d_in[7];

  const int N = in_sizes[0] / DIM;   // 100000 (divisible by 16)
  const int E = in_sizes[1] / 2;     // 1600000
  const int* src = edge;
  const int* dst = edge + E;

  float* agg = (float*)d_ws;                 // N*128 f32
  float* cnt = agg + (size_t)N * DIM;        // N f32
  float* h   = cnt + N;                      // N*128 f32
  float* out = (float*)d_out;

  const int edges_per_block = 256 / 32;
  const int scatter_blocks  = (E + edges_per_block - 1) / edges_per_block;
  const int gemm_blocks     = N / 16;

  // ----- layer 1 -----
  sage_zero_f32<<<2048, 256, 0, stream>>>(agg, (long long)N * DIM + N); // agg + cnt
  sage_degree<<<2048, 256, 0, stream>>>(dst, cnt, E);
  sage_scatter<<<scatter_blocks, 256, 0, stream>>>(x, src, dst, agg, E);
  sage_fused_gemm<<<gemm_blocks, 256, 0, stream>>>(agg, cnt, x, W1l, W1r, b1, h, 1);

  // ----- layer 2 (degrees unchanged; reuse cnt) -----
  sage_zero_f32<<<2048, 256, 0, stream>>>(agg, (long long)N * DIM);
  sage_scatter<<<scatter_blocks, 256, 0, stream>>>(h, src, dst, agg, E);
  sage_fused_gemm<<<gemm_blocks, 256, 0, stream>>>(agg, cnt, h, W2l, W2r, b2, out, 0);
}